// TorchKernelRegression_39908836115039
// MI455X (gfx1250) — compile-verified
//
#include <hip/hip_runtime.h>

typedef __attribute__((ext_vector_type(2))) float v2f;
typedef __attribute__((ext_vector_type(8))) float v8f;

#define CHUNK 4096          // floats per chunk per array (16 KB each)
#define BW    0.01f

// Low 32 bits of a generic pointer to __shared__ = LDS byte offset
// (LDS aperture maps addr[31:0] directly to LDS space).
__device__ __forceinline__ unsigned lds_off(const void* p) {
    return (unsigned)(unsigned long long)(size_t)p;
}

// Async-copy this wave's 512-float slice of one chunk: 8 x b64 per lane
// (32 lanes x 8 B x 8 insts = 2 KB/wave x 8 waves... per array slice).
// Tracked on ASYNCcnt; GV addressing (64-bit vaddr, saddr=off).
__device__ __forceinline__ void async_copy_slice(float* lds_dst, const float* gsrc,
                                                 int wave, int lane) {
    const int pair = wave * 256 + lane;            // b64 pair index within chunk
    unsigned l = lds_off(lds_dst) + (unsigned)(pair * 8);
    unsigned long long g = (unsigned long long)(size_t)gsrc +
                           (unsigned long long)(pair * 8);
    #pragma unroll
    for (int i = 0; i < 8; ++i) {
        asm volatile("global_load_async_to_lds_b64 %0, %1, off"
                     :: "v"(l), "v"(g) : "memory");
        l += 256;            // 32 lanes * 8 B
        g += 256;
    }
}

// Each wave owns 16 queries (rows) and sweeps memory points in 16-wide tiles
// (columns). Diff tile on the matrix pipe:
//   D = A(16x4) x B(4x16), A=[x_q, 1, 0, 0], B=[[1..],[-x_mem..],[0],[0]]
// => D[q,n] = x_q - x_n bit-exactly in f32 (matches reference's f32 subtract).
__global__ __launch_bounds__(256) void kreg_box_wmma(
    const float* __restrict__ x_mem,
    const float* __restrict__ y_mem,
    const float* __restrict__ x_query,
    float* __restrict__ out,
    int n_mem)
{
    __shared__ float xs[2][CHUNK];   // double-buffered x_mem chunks
    __shared__ float ys[2][CHUNK];   // double-buffered y_mem chunks

    const int tid  = threadIdx.x;
    const int lane = tid & 31;
    const int wave = tid >> 5;
    const bool lo  = lane < 16;
    const int col  = lane & 15;
    const int q_base = (blockIdx.x * 8 + wave) * 16;

    // A operand (loop-invariant):
    // lanes 0-15:  VGPR0 = K0 = x_query[row], VGPR1 = K1 = 1.0
    // lanes 16-31: VGPR0 = K2 = 0,            VGPR1 = K3 = 0
    v2f a;
    a.x = lo ? x_query[q_base + lane] : 0.0f;
    a.y = lo ? 1.0f : 0.0f;

    v8f num = {};
    v8f den = {};

    const int nchunks = n_mem / CHUNK;

    // prime the pipeline: chunk 0 -> buffer 0 (16 async b64 per wave)
    async_copy_slice(xs[0], x_mem, wave, lane);
    async_copy_slice(ys[0], y_mem, wave, lane);

    for (int c = 0; c < nchunks; ++c) {
        const int cb = c & 1;

        if (c + 1 < nchunks) {
            // prefetch next chunk into the other buffer (safe: that buffer's
            // previous consumers all passed the end-of-iteration barrier)
            async_copy_slice(xs[cb ^ 1], x_mem + (c + 1) * CHUNK, wave, lane);
            async_copy_slice(ys[cb ^ 1], y_mem + (c + 1) * CHUNK, wave, lane);
            // 16 newer loads outstanding -> waiting to <=16 retires chunk c
            asm volatile("s_wait_asynccnt 0x10" ::: "memory");
        } else {
            asm volatile("s_wait_asynccnt 0x0" ::: "memory");
        }
        __syncthreads();   // all waves' slices of chunk c resident

        const float* __restrict__ xc = xs[cb];
        const float* __restrict__ yc = ys[cb];

        #pragma unroll 2
        for (int nb = 0; nb < CHUNK; nb += 16) {
            const float xv = xc[nb + col];   // lanes L and L+16 broadcast-share
            const float yv = yc[nb + col];

            // B operand:
            // VGPR0: lanes 0-15 = row K0 (ones), lanes 16-31 = row K1 (-x_mem)
            // VGPR1: rows K2,K3 = 0
            v2f b;
            b.x = lo ? 1.0f : (0.0f - xv);
            b.y = 0.0f;

            v8f cacc = {};
            v8f d = __builtin_amdgcn_wmma_f32_16x16x4_f32(
                false, a, false, b, (short)0, cacc, false, false);

            #pragma unroll
            for (int v = 0; v < 8; ++v) {
                float w = (__builtin_fabsf(d[v]) <= BW) ? 1.0f : 0.0f;
                num[v] = __builtin_fmaf(w, yv, num[v]);
                den[v] += w;
            }
        }
        __syncthreads();   // done reading buffer cb before it is overwritten
    }

    // Reduce across the 16 mem-columns held by each 16-lane half
    // (xor masks 1,2,4,8 never cross the half boundary, bit 4 untouched).
    #pragma unroll
    for (int m = 1; m <= 8; m <<= 1) {
        #pragma unroll
        for (int v = 0; v < 8; ++v) {
            num[v] += __shfl_xor(num[v], m, 32);
            den[v] += __shfl_xor(den[v], m, 32);
        }
    }

    // lane 0 holds queries q_base+0..7 (rows M=0..7),
    // lane 16 holds queries q_base+8..15 (rows M=8..15)
    if (lane == 0 || lane == 16) {
        const int qo = q_base + (lo ? 0 : 8);
        #pragma unroll
        for (int v = 0; v < 8; ++v) {
            out[qo + v] = num[v] / den[v];
        }
    }
}

extern "C" void kernel_launch(void* const* d_in, const int* in_sizes, int n_in,
                              void* d_out, int out_size, void* d_ws, size_t ws_size,
                              hipStream_t stream) {
    const float* x_mem   = (const float*)d_in[0];
    const float* y_mem   = (const float*)d_in[1];
    const float* x_query = (const float*)d_in[2];
    float* out = (float*)d_out;

    const int n_mem = in_sizes[0];   // 16384
    const int m_q   = out_size;      // 16384

    // 256 threads = 8 waves = 128 queries per block
    dim3 block(256);
    dim3 grid(m_q / 128);
    kreg_box_wmma<<<grid, block, 0, stream>>>(x_mem, y_mem, x_query, out, n_mem);
}